// ModelNew_3556232922006
// MI455X (gfx1250) — compile-verified
//
#include <hip/hip_runtime.h>

typedef __attribute__((ext_vector_type(2))) float v2f;
typedef __attribute__((ext_vector_type(8))) float v8f;

#define BATCH 8
#define CIN   16
#define COUT  32
#define DIN   24
#define POUT  11
#define XC_STRIDE (DIN*DIN*DIN)      // 13824 floats between ci planes
#define NW_ELEMS (27*4*32*2)         // 6912 float2 = 55296 B of LDS

// One block (64 threads = 2 waves) computes one pooled output voxel (n,pd,ph,pw)
// for all 32 output channels; wave w covers couts [16w,16w+16).
// ConvTranspose3d as implicit GEMM on V_WMMA_F32_16X16X4_F32 with BN affine and
// the 4x4x4 double-avgpool fused via cross-lane reduction of the C tile.
// Weights are staged once into LDS in the exact A-operand layout so each WMMA
// fetches A with a single conflict-free ds_load_b64; x tap values are pure
// parity selects over an unconditionally-loaded 2x2 register patch (no
// predicated loads, EXEC stays all-ones).
__global__ __launch_bounds__(64)
void convt3d_bn_pool_wmma(const float* __restrict__ x,
                          const float* __restrict__ weight,
                          const float* __restrict__ bias,
                          const float* __restrict__ gamma,
                          const float* __restrict__ beta,
                          const float* __restrict__ rmean,
                          const float* __restrict__ rvar,
                          float* __restrict__ out) {
    __shared__ v2f wlds[NW_ELEMS];   // [tap][q][co][half] -> (w[ci0], w[ci0+1])

    const int tid    = threadIdx.x;
    const int lane   = tid & 31;
    const int waveId = tid >> 5;
    const int co_base = waveId * 16;
    const int nl   = lane & 15;      // N column (B/C), M row (A) within tile
    const int half = lane >> 4;      // which K-pair this lane holds

    // ---- stage all weights to LDS, transposed into A-operand layout --------
    for (int i = tid; i < NW_ELEMS; i += 64) {
        const int h   = i & 1;
        const int co  = (i >> 1) & 31;
        const int q   = (i >> 6) & 3;
        const int tap = i >> 8;                  // kd*9 + kh*3 + kw
        const int ci0 = 4 * q + 2 * h;
        v2f w2;
        w2.x = weight[(ci0       * COUT + co) * 27 + tap];
        w2.y = weight[((ci0 + 1) * COUT + co) * 27 + tap];
        wlds[i] = w2;
    }
    __syncthreads();

    int bid = blockIdx.x;
    const int pw = bid % POUT; bid /= POUT;
    const int ph = bid % POUT; bid /= POUT;
    const int pd = bid % POUT; bid /= POUT;
    const int n  = bid;

    // conv-output coords covered by this lane's N column (4x4 hw patch)
    const int oh = 4 * ph + (nl >> 2);
    const int ow = 4 * pw + (nl & 3);
    // candidate input rows/cols (always in [0,24) -- no bounds masking needed)
    const int r0 = oh >> 1, r1 = (oh + 1) >> 1;
    const int c0 = ow >> 1, c1 = (ow + 1) >> 1;
    const bool hodd = (oh & 1) != 0;
    const bool wodd = (ow & 1) != 0;

    const float* xn = x + (size_t)n * CIN * XC_STRIDE;

    v8f acc = {};   // 16x16 C tile: rows = cout, cols = hw patch, summed over od

    // od = 2*id + kd - 1 must lie in [4pd, 4pd+4)  (id = 2pd + t)
    const int kdlo[3] = {1, 0, 0};
    const int kdhi[3] = {2, 2, 0};

#pragma unroll
    for (int t = 0; t < 3; ++t) {
        const float* xslice = xn + (2 * pd + t) * (DIN * DIN);
        for (int q = 0; q < 4; ++q) {
            const int ci0 = 4 * q + 2 * half;
            // unconditional 2x2 input patch per ci component
            float g[2][2][2];
#pragma unroll
            for (int e = 0; e < 2; ++e) {
                const float* xb = xslice + (ci0 + e) * XC_STRIDE;
                g[e][0][0] = xb[r0 * DIN + c0];
                g[e][0][1] = xb[r0 * DIN + c1];
                g[e][1][0] = xb[r1 * DIN + c0];
                g[e][1][1] = xb[r1 * DIN + c1];
            }
            // 3x3 tap values per component: parity selects only (v_cndmask)
            // kh=0 -> row r1, valid if oh odd; kh=1 -> r0, even; kh=2 -> r0, odd
            // kw=0 -> col c1, valid if ow odd; kw=1 -> c0, even; kw=2 -> c0, odd
            float tv[2][3][3];
#pragma unroll
            for (int e = 0; e < 2; ++e) {
                const float h0c0 = hodd ? g[e][1][0] : 0.f;
                const float h0c1 = hodd ? g[e][1][1] : 0.f;
                const float h1c0 = hodd ? 0.f : g[e][0][0];
                const float h1c1 = hodd ? 0.f : g[e][0][1];
                const float h2c0 = hodd ? g[e][0][0] : 0.f;
                const float h2c1 = hodd ? g[e][0][1] : 0.f;
                tv[e][0][0] = wodd ? h0c1 : 0.f;
                tv[e][0][1] = wodd ? 0.f : h0c0;
                tv[e][0][2] = wodd ? h0c0 : 0.f;
                tv[e][1][0] = wodd ? h1c1 : 0.f;
                tv[e][1][1] = wodd ? 0.f : h1c0;
                tv[e][1][2] = wodd ? h1c0 : 0.f;
                tv[e][2][0] = wodd ? h2c1 : 0.f;
                tv[e][2][1] = wodd ? 0.f : h2c0;
                tv[e][2][2] = wodd ? h2c0 : 0.f;
            }
#pragma unroll
            for (int kd = kdlo[t]; kd <= kdhi[t]; ++kd) {
#pragma unroll
                for (int kh = 0; kh < 3; ++kh) {
#pragma unroll
                    for (int kw = 0; kw < 3; ++kw) {
                        const int tap = kd * 9 + kh * 3 + kw;
                        // A tile: one conflict-free ds_load_b64 per WMMA
                        v2f a = wlds[((tap * 4 + q) * 32 + (co_base + nl)) * 2 + half];
                        v2f b;
                        b.x = tv[0][kh][kw];
                        b.y = tv[1][kh][kw];
                        acc = __builtin_amdgcn_wmma_f32_16x16x4_f32(
                                false, a, false, b, (short)0, acc, false, false);
                    }
                }
            }
        }
    }

    // ---- fused double-avgpool + BatchNorm affine epilogue -------------------
    // Sum the C tile over its 16 N columns (each 16-lane half holds 8 cout rows)
    const size_t out_nbase = ((size_t)n * COUT) * (POUT * POUT * POUT)
                           + (size_t)pd * (POUT * POUT) + ph * POUT + pw;
#pragma unroll
    for (int j = 0; j < 8; ++j) {
        float v = acc[j];
        v += __shfl_xor(v, 8, 32);
        v += __shfl_xor(v, 4, 32);
        v += __shfl_xor(v, 2, 32);
        v += __shfl_xor(v, 1, 32);
        if (nl == 0) {
            const int co = co_base + j + 8 * half;  // lane0 -> row j, lane16 -> j+8
            const float sc  = gamma[co] * rsqrtf(rvar[co] + 1e-5f);
            const float sh  = beta[co] - rmean[co] * sc;
            const float val = (v * (1.0f / 64.0f) + bias[co]) * sc + sh;
            out[out_nbase + (size_t)co * (POUT * POUT * POUT)] = val;
        }
    }
}

extern "C" void kernel_launch(void* const* d_in, const int* in_sizes, int n_in,
                              void* d_out, int out_size, void* d_ws, size_t ws_size,
                              hipStream_t stream) {
    const float* x      = (const float*)d_in[0];
    const float* weight = (const float*)d_in[1];
    const float* bias   = (const float*)d_in[2];
    const float* gamma  = (const float*)d_in[3];
    const float* beta   = (const float*)d_in[4];
    const float* rmean  = (const float*)d_in[5];
    const float* rvar   = (const float*)d_in[6];
    float* out = (float*)d_out;

    const int nblocks = BATCH * POUT * POUT * POUT;  // 10648 pooled voxels
    convt3d_bn_pool_wmma<<<nblocks, 64, 0, stream>>>(
        x, weight, bias, gamma, beta, rmean, rvar, out);
}